// CRY_50259707298076
// MI455X (gfx1250) — compile-verified
//
#include <hip/hip_runtime.h>

typedef __attribute__((ext_vector_type(2))) float v2f;
typedef __attribute__((ext_vector_type(8))) float v8f;

namespace {
constexpr int D = 4096;           // 2^12 states
constexpr int B = 128;            // batch columns
constexpr int WAVES_PER_BLOCK = 8;
constexpr int TILES = (D / 16) * (B / 16);  // 256 row-tiles * 8 col-tiles = 2048
}

// BD 4x4 block entry: rows/cols 0..3 over (bit11,bit10) combos.
// [1 0 0 0; 0 1 0 0; 0 0 c -s; 0 0 -s c]
__device__ __forceinline__ float m4(int r, int col, float c, float s) {
  return (r == col) ? ((r < 2) ? 1.0f : c)
                    : ((r + col == 5) ? -s : 0.0f);
}

__global__ __launch_bounds__(256) void cry_wmma_kernel(
    const float* __restrict__ xr, const float* __restrict__ xi,
    const float* __restrict__ theta, float* __restrict__ out) {
  const int lane = threadIdx.x & 31;
  const int wave = threadIdx.x >> 5;
  const int tile = blockIdx.x * WAVES_PER_BLOCK + wave;  // 0..2047
  const int rowTile = tile >> 3;                         // 0..255
  const int colTile = tile & 7;                          // 0..7
  const int rtb = rowTile << 2;          // base of low-10-bit row group, 0..1020
  const int col = (colTile << 4) + (lane & 15);          // batch column 0..127

  // theta/2 is a few radians at most (theta ~ pi*N(0,1)); native
  // v_cos_f32/v_sin_f32 are accurate to ~1 ulp-ish here and avoid the
  // ~350-instruction ocml range-reduction per wave.
  const float th = 0.5f * theta[0];
  const float cth = __cosf(th);
  const float sth = __sinf(th);

  const bool lower = lane < 16;
  const int rowm = lane & 15;            // A-matrix row M held by this lane
  const int kbase = lower ? 0 : 2;       // K indices {kbase, kbase+1} in this lane

  v8f accR = {};
  v8f accI = {};

#pragma unroll
  for (int k = 0; k < 4; ++k) {
    // A_k = columns 4k..4k+3 of BD16: nonzero only for row-group == k.
    const bool hit = ((rowm >> 2) == k);
    v2f a;
    a.x = hit ? m4(rowm & 3, kbase + 0, cth, sth) : 0.0f;
    a.y = hit ? m4(rowm & 3, kbase + 1, cth, sth) : 0.0f;

    // B_k rows (K index j) = state rows (rtb + k) + j*1024, 16 batch cols.
    const int xrow = rtb + k;
    const size_t i0 = (size_t)(xrow + (kbase + 0) * 1024) * B + col;
    const size_t i1 = (size_t)(xrow + (kbase + 1) * 1024) * B + col;
    v2f br, bi;
    br.x = xr[i0]; br.y = xr[i1];
    bi.x = xi[i0]; bi.y = xi[i1];

    accR = __builtin_amdgcn_wmma_f32_16x16x4_f32(
        false, a, false, br, (short)0, accR, false, false);
    accI = __builtin_amdgcn_wmma_f32_16x16x4_f32(
        false, a, false, bi, (short)0, accI, false, false);
  }

  // D 16x16 layout: VGPR j -> M=j (lanes 0-15) / M=j+8 (lanes 16-31).
  // Row m of BD-output: group g=m>>2 (low bits rtb+g), combo c=m&3 (bits 11:10).
#pragma unroll
  for (int j = 0; j < 8; ++j) {
    const int m = lower ? j : (j + 8);
    const int grow = rtb + (m >> 2) + (m & 3) * 1024;  // global state row
    float2 v;
    v.x = accR[j];
    v.y = accI[j];
    *reinterpret_cast<float2*>(out + 2 * ((size_t)grow * B + col)) = v;
  }
}

extern "C" void kernel_launch(void* const* d_in, const int* in_sizes, int n_in,
                              void* d_out, int out_size, void* d_ws, size_t ws_size,
                              hipStream_t stream) {
  const float* xr = (const float*)d_in[0];   // x_real (D, B) f32
  const float* xi = (const float*)d_in[1];   // x_imag (D, B) f32
  const float* th = (const float*)d_in[2];   // theta (1,)
  float* out = (float*)d_out;                // complex64 (D, B) interleaved

  dim3 grid(TILES / WAVES_PER_BLOCK);  // 256 blocks
  dim3 block(32 * WAVES_PER_BLOCK);    // 256 threads = 8 wave32 tiles
  hipLaunchKernelGGL(cry_wmma_kernel, grid, block, 0, stream, xr, xi, th, out);
}